// get_model_55662776156613
// MI455X (gfx1250) — compile-verified
//
#include <hip/hip_runtime.h>

typedef float v2f __attribute__((ext_vector_type(2)));
typedef float v8f __attribute__((ext_vector_type(8)));

#define EPSV 1e-5f
#define BN_CHUNK 65536LL

static inline int cdiv(long long a, long long b){ return (int)((a + b - 1)/b); }
static inline int rup4(int c){ return (c + 3) & ~3; }

#define WMMA_F32X4(a, b, c) \
  __builtin_amdgcn_wmma_f32_16x16x4_f32(false, (a), false, (b), (short)0, (c), false, false)

// ---------------- transpose (B,3,N) -> (B,N,3) ----------------
__global__ void transpose_k(const float* __restrict__ x, float* __restrict__ y, int B, int N){
  int i = blockIdx.x*blockDim.x + threadIdx.x;
  if (i >= B*N) return;
  int b = i / N, n = i % N;
#pragma unroll
  for (int c = 0; c < 3; ++c)
    y[((long long)b*N + n)*3 + c] = x[((long long)b*3 + c)*N + n];
}

// ---------------- farthest point sampling: one block per batch ----------------
__global__ void fps_k(const float* __restrict__ xyz, int N, int npoint, int* __restrict__ out){
  __shared__ float dist[2048];
  __shared__ float rv[256];
  __shared__ int   ri[256];
  __shared__ int   s_far;
  int b = blockIdx.x, t = threadIdx.x;
  const float* p = xyz + (long long)b*N*3;
  for (int i = t; i < N; i += blockDim.x) dist[i] = 1e10f;
  if (t == 0) s_far = 0;
  __syncthreads();
  for (int s = 0; s < npoint; ++s) {
    int far = s_far;
    if (t == 0) out[b*npoint + s] = far;
    float cx = p[far*3+0], cy = p[far*3+1], cz = p[far*3+2];
    float best = -1.f; int bi = 0x7fffffff;
    for (int i = t; i < N; i += blockDim.x) {
      float dx = p[i*3+0]-cx, dy = p[i*3+1]-cy, dz = p[i*3+2]-cz;
      float d = dx*dx + dy*dy + dz*dz;
      float nd = fminf(dist[i], d);
      dist[i] = nd;
      if (nd > best) { best = nd; bi = i; }
    }
    rv[t] = best; ri[t] = bi;
    __syncthreads();
    for (int off = blockDim.x>>1; off > 0; off >>= 1) {
      if (t < off) {
        bool take = (rv[t+off] > rv[t]) || (rv[t+off] == rv[t] && ri[t+off] < ri[t]);
        if (take) { rv[t] = rv[t+off]; ri[t] = ri[t+off]; }
      }
      __syncthreads();
    }
    if (t == 0) s_far = ri[0];
    __syncthreads();
  }
}

// ---------------- gather sampled centers ----------------
__global__ void gather_xyz_k(const float* __restrict__ xyz, const int* __restrict__ idx,
                             float* __restrict__ out, int total, int S, int N){
  int i = blockIdx.x*blockDim.x + threadIdx.x;
  if (i >= total) return;
  int b = i / S;
  int id = idx[i];
#pragma unroll
  for (int c = 0; c < 3; ++c)
    out[(long long)i*3 + c] = xyz[((long long)b*N + id)*3 + c];
}

// ---------------- ball query: first K indices (ascending) within radius ----------------
__global__ void ball_query_k(const float* __restrict__ xyz, const float* __restrict__ nxyz,
                             int B, int N, int S, float r2, int K, int* __restrict__ idx){
  int i = blockIdx.x*blockDim.x + threadIdx.x;
  if (i >= B*S) return;
  int b = i / S;
  const float* p = xyz + (long long)b*N*3;
  float qx = nxyz[(long long)i*3+0], qy = nxyz[(long long)i*3+1], qz = nxyz[(long long)i*3+2];
  int* o = idx + (long long)i*K;
  int cnt = 0, first = 0;
  for (int n = 0; n < N; ++n) {
    float dx = p[n*3+0]-qx, dy = p[n*3+1]-qy, dz = p[n*3+2]-qz;
    float d = dx*dx + dy*dy + dz*dz;
    if (d <= r2) {
      if (cnt == 0) first = n;
      o[cnt++] = n;
      if (cnt == K) break;
    }
  }
  for (; cnt < K; ++cnt) o[cnt] = first;
}

// ---------------- grouping: [features(Cf) | xyz-center(3) | zero-pad] stride C4 ----------------
__global__ void group_k(const float* __restrict__ xyz, const float* __restrict__ nxyz,
                        const float* __restrict__ pts, const int* __restrict__ idx,
                        int N, int S, int K, int Cf, int C4, float* __restrict__ g, int total){
  int m = blockIdx.x*blockDim.x + threadIdx.x;
  if (m >= total) return;
  int bs = m / K;            // b*S + s
  int b = bs / S;
  int id = idx[m];
  float* gr = g + (long long)m*C4;
  if (pts) {
    const float* ps = pts + ((long long)b*N + id)*Cf;
    for (int c = 0; c < Cf; ++c) gr[c] = ps[c];
  }
  const float* px = xyz + ((long long)b*N + id)*3;
  const float* q  = nxyz + (long long)bs*3;
  gr[Cf+0] = px[0]-q[0]; gr[Cf+1] = px[1]-q[1]; gr[Cf+2] = px[2]-q[2];
  for (int c = Cf+3; c < C4; ++c) gr[c] = 0.f;
}

// ---------------- pack weights (O,C) -> zero-padded (O,C4) ----------------
__global__ void pack_w_k(const float* __restrict__ W, float* __restrict__ Wp,
                         int O, int C, int C4){
  int i = blockIdx.x*blockDim.x + threadIdx.x;
  if (i >= O*C4) return;
  int o = i / C4, c = i % C4;
  Wp[i] = (c < C) ? W[(long long)o*C + c] : 0.f;
}

// ---------------- fp32 WMMA GEMM: out[m,o] = sum_c A[m,c]*Wp[o,c] + bias[o] ----------------
// One wave computes a 32x32 output tile: 2 M-fragments x 2 N-fragments, 4 WMMA
// accumulators, so each K-step is 4 float2 loads feeding 4 v_wmma_f32_16x16x4_f32
// (1:1 load:wmma, both fragments reused twice). Operands use padded stride C4
// (multiple of 4) -> branch-free 8B-aligned loads. The loop is software-pipelined:
// fragments for step i+1 are fetched before the step-i WMMAs consume the current ones.
// A 16x4 layout: lane(0-15) v0/v1 = K0/K1, lane(16-31) = K2/K3 (M = lane%16);
// B 4x16 mirrors with N = lane%16; D: acc[j] = D[M = j + 8*(lane>=16)][N = lane%16].
__global__ void wmma_gemm_k(const float* __restrict__ A, const float* __restrict__ Wp,
                            const float* __restrict__ bias, float* __restrict__ out,
                            long long M, int C4, int O){
  int lane = threadIdx.x & 31;
  int wave = threadIdx.x >> 5;
  long long tile   = (long long)blockIdx.x * (blockDim.x >> 5) + wave;
  long long tilesN = (long long)(O >> 5);      // 32-wide N tiles
  long long tilesM = M >> 5;                   // 32-tall M tiles
  if (tile >= tilesM * tilesN) return;
  int       tn = (int)(tile % tilesN);
  long long tm = tile / tilesN;
  int l16  = lane & 15;
  int half = lane >> 4;                         // 0: K0/K1, 1: K2/K3
  const float* Ar0 = A  + (tm*32 + l16) * (long long)C4 + half*2;
  const float* Ar1 = Ar0 + (long long)16 * C4;
  const float* Wr0 = Wp + ((long long)(tn*32 + l16)) * C4 + half*2;
  const float* Wr1 = Wr0 + (long long)16 * C4;
  v8f acc00 = {}, acc01 = {}, acc10 = {}, acc11 = {};
  v2f a0 = *(const v2f*)(Ar0);
  v2f a1 = *(const v2f*)(Ar1);
  v2f b0 = *(const v2f*)(Wr0);
  v2f b1 = *(const v2f*)(Wr1);
  for (int k0 = 4; k0 < C4; k0 += 4) {
    v2f na0 = *(const v2f*)(Ar0 + k0);
    v2f na1 = *(const v2f*)(Ar1 + k0);
    v2f nb0 = *(const v2f*)(Wr0 + k0);
    v2f nb1 = *(const v2f*)(Wr1 + k0);
    acc00 = WMMA_F32X4(a0, b0, acc00);
    acc01 = WMMA_F32X4(a0, b1, acc01);
    acc10 = WMMA_F32X4(a1, b0, acc10);
    acc11 = WMMA_F32X4(a1, b1, acc11);
    a0 = na0; a1 = na1; b0 = nb0; b1 = nb1;
  }
  acc00 = WMMA_F32X4(a0, b0, acc00);
  acc01 = WMMA_F32X4(a0, b1, acc01);
  acc10 = WMMA_F32X4(a1, b0, acc10);
  acc11 = WMMA_F32X4(a1, b1, acc11);

  int oc = tn*32 + l16;
  long long mb0 = tm*32 + 8*half;
  long long mb1 = mb0 + 16;
  float bb0 = bias[oc], bb1 = bias[oc + 16];
#pragma unroll
  for (int j = 0; j < 8; ++j) {
    out[(mb0 + j)*O + oc]      = acc00[j] + bb0;
    out[(mb0 + j)*O + oc + 16] = acc01[j] + bb1;
    out[(mb1 + j)*O + oc]      = acc10[j] + bb0;
    out[(mb1 + j)*O + oc + 16] = acc11[j] + bb1;
  }
}

// ---------------- batchnorm statistics: sums[o], sums[O+o] over M rows ----------------
__global__ void bn_stats_k(const float* __restrict__ x, long long M, int O, float* __restrict__ sums){
  __shared__ float rs[256], rq[256];
  int o = blockIdx.x;
  long long start = (long long)blockIdx.y * BN_CHUNK;
  long long end = start + BN_CHUNK; if (end > M) end = M;
  float s = 0.f, q = 0.f;
  for (long long m = start + threadIdx.x; m < end; m += blockDim.x) {
    float v = x[m*O + o];
    s += v; q += v*v;
  }
  rs[threadIdx.x] = s; rq[threadIdx.x] = q;
  __syncthreads();
  for (int off = 128; off > 0; off >>= 1) {
    if ((int)threadIdx.x < off) { rs[threadIdx.x] += rs[threadIdx.x+off]; rq[threadIdx.x] += rq[threadIdx.x+off]; }
    __syncthreads();
  }
  if (threadIdx.x == 0) { atomicAdd(&sums[o], rs[0]); atomicAdd(&sums[O+o], rq[0]); }
}

__global__ void bn_relu_k(float* __restrict__ x, long long M, int O,
                          const float* __restrict__ sums,
                          const float* __restrict__ gm, const float* __restrict__ bt){
  long long i = (long long)blockIdx.x*blockDim.x + threadIdx.x;
  if (i >= M*O) return;
  int o = (int)(i % O);
  float inv = 1.f/(float)M;
  float mean = sums[o]*inv;
  float var  = sums[O+o]*inv - mean*mean;
  float v = gm[o]*(x[i]-mean)*rsqrtf(var + EPSV) + bt[o];
  x[i] = fmaxf(v, 0.f);
}

// ---------------- max over K with concat write ----------------
__global__ void maxpool_k(const float* __restrict__ x, int T /*B*S*/, int K, int O,
                          float* __restrict__ out, int Ctot, int co){
  int i = blockIdx.x*blockDim.x + threadIdx.x;
  if (i >= T*O) return;
  int o = i % O, bs = i / O;
  float m = -3.4e38f;
  const float* base = x + ((long long)bs*K)*O + o;
  for (int k = 0; k < K; ++k) m = fmaxf(m, base[(long long)k*O]);
  out[(long long)bs*Ctot + co + o] = m;
}

// ---------------- SA3 input: [l2_xyz(3) | l2_pts(640) | pad] stride 644 ----------------
__global__ void build_sa3_k(const float* __restrict__ l2xyz, const float* __restrict__ l2pts,
                            float* __restrict__ g, int total){
  int m = blockIdx.x*blockDim.x + threadIdx.x;
  if (m >= total) return;
  float* gr = g + (long long)m*644;
  gr[0] = l2xyz[m*3+0]; gr[1] = l2xyz[m*3+1]; gr[2] = l2xyz[m*3+2];
  const float* ps = l2pts + (long long)m*640;
  for (int c = 0; c < 640; ++c) gr[3+c] = ps[c];
  gr[643] = 0.f;
}

// ---------------- tiny FC (M=8) and BN-over-batch ----------------
__global__ void fc_k(const float* __restrict__ x, const float* __restrict__ W,
                     const float* __restrict__ b, float* __restrict__ out,
                     int M, int C, int O){
  int i = blockIdx.x*blockDim.x + threadIdx.x;
  if (i >= M*O) return;
  int o = i % O, m = i / O;
  const float* xr = x + (long long)m*C;
  const float* wr = W + (long long)o*C;
  float s = b[o];
  for (int c = 0; c < C; ++c) s += xr[c]*wr[c];
  out[i] = s;
}

__global__ void bn1d_relu_k(const float* __restrict__ src, float* __restrict__ dst,
                            int M, int O, const float* __restrict__ gm, const float* __restrict__ bt){
  int o = blockIdx.x*blockDim.x + threadIdx.x;
  if (o >= O) return;
  float s = 0.f, q = 0.f;
  for (int m = 0; m < M; ++m) { float v = src[(long long)m*O+o]; s += v; q += v*v; }
  float mean = s/(float)M, var = q/(float)M - mean*mean;
  float inv = rsqrtf(var + EPSV);
  for (int m = 0; m < M; ++m) {
    float v = gm[o]*(src[(long long)m*O+o]-mean)*inv + bt[o];
    dst[(long long)m*O+o] = fmaxf(v, 0.f);
  }
}

// ================= host orchestration =================
extern "C" void kernel_launch(void* const* d_in, const int* in_sizes, int n_in,
                              void* d_out, int out_size, void* d_ws, size_t ws_size,
                              hipStream_t stream) {
  (void)in_sizes; (void)n_in; (void)out_size; (void)ws_size;
  auto P = [&](int i){ return (const float*)d_in[i]; };

  // ---- workspace carve-out (floats) ----
  float* w = (float*)d_ws;
  size_t off = 0;
  auto alloc = [&](size_t nf){ float* p = w + off; off += nf; return p; };
  float* xyz_t = alloc((size_t)8*2048*3);
  int*   fidx1 = (int*)alloc(8*512);
  float* nxyz1 = alloc((size_t)8*512*3);
  float* l1pts = alloc((size_t)8*512*320);
  int*   fidx2 = (int*)alloc(8*128);
  float* nxyz2 = alloc((size_t)8*128*3);
  float* l2pts = alloc((size_t)8*128*640);
  int*   idxbuf= (int*)alloc((size_t)8*512*128);
  float* gbuf  = alloc((size_t)131072*324);          // max grouped tensor (padded stride)
  float* acta  = alloc((size_t)33554432);            // SA2 max rows*O (SA1 K=128 branch needs 2x)
  float* actb  = alloc((size_t)33554432);
  float* wpad  = alloc((size_t)1024*644);            // padded weights (max O*C4)
  float* sums  = alloc(2048);                        // [O] sum + [O] sumsq
  float* xpool = alloc((size_t)8*1024);
  float* h1    = alloc((size_t)8*512);
  float* t2    = alloc((size_t)8*256);
  float* h2    = alloc((size_t)8*256);

  const float* xyz_in = (const float*)d_in[0];
  transpose_k<<<cdiv(8*2048,256),256,0,stream>>>(xyz_in, xyz_t, 8, 2048);

  auto launch_gemm = [&](const float* A, const float* Worig, const float* bias,
                         float* out, long long M, int C, int C4, int O){
    pack_w_k<<<cdiv((long long)O*C4,256),256,0,stream>>>(Worig, wpad, O, C, C4);
    long long tiles = (M/32) * (long long)(O/32);
    int blocks = cdiv(tiles, 8);                     // 8 waves per 256-thread block
    wmma_gemm_k<<<blocks, 256, 0, stream>>>(A, wpad, bias, out, M, C4, O);
  };

  auto run_layers = [&](const float* in0, long long Mrows, int Cin0,
                        const int* mlp, int pbase, float** lastOut, int* lastO){
    const float* in = in0;
    int Cin = Cin0, lda = rup4(Cin0);
    float* bufs[2] = {acta, actb};
    for (int l = 0; l < 3; ++l) {
      int O = mlp[l];
      float* ob = bufs[l & 1];
      launch_gemm(in, P(pbase + l*4 + 0), P(pbase + l*4 + 1), ob, Mrows, Cin, lda, O);
      hipMemsetAsync(sums, 0, (size_t)2*O*sizeof(float), stream);
      dim3 gs((unsigned)O, (unsigned)((Mrows + BN_CHUNK - 1)/BN_CHUNK));
      bn_stats_k<<<gs, 256, 0, stream>>>(ob, Mrows, O, sums);
      bn_relu_k<<<cdiv(Mrows*O,256), 256, 0, stream>>>(ob, Mrows, O, sums,
                                                       P(pbase + l*4 + 2), P(pbase + l*4 + 3));
      in = ob; Cin = O; lda = O;                     // O is a multiple of 4
      *lastOut = ob; *lastO = O;
    }
  };

  auto run_sa = [&](const float* xyzL, int N, const float* pts, int Cf, int npoint,
                    const float* radii, const int* ks, const int (*mlps)[3],
                    int pbase, int* fidx, float* nxyz, float* outpts, int Ctot){
    fps_k<<<8, 256, 0, stream>>>(xyzL, N, npoint, fidx);
    int tot = 8*npoint;
    gather_xyz_k<<<cdiv(tot,256),256,0,stream>>>(xyzL, fidx, nxyz, tot, npoint, N);
    int co = 0;
    for (int br = 0; br < 3; ++br) {
      int K = ks[br];
      float r2 = radii[br]*radii[br];
      ball_query_k<<<cdiv(tot,256),256,0,stream>>>(xyzL, nxyz, 8, N, npoint, r2, K, idxbuf);
      long long Mrows = (long long)tot*K;
      int C4 = rup4(Cf + 3);
      group_k<<<cdiv(Mrows,256),256,0,stream>>>(xyzL, nxyz, pts, idxbuf, N, npoint, K, Cf, C4,
                                                gbuf, (int)Mrows);
      float* last = nullptr; int lastO = 0;
      run_layers(gbuf, Mrows, Cf+3, mlps[br], pbase + br*12, &last, &lastO);
      maxpool_k<<<cdiv((long long)tot*lastO,256),256,0,stream>>>(last, tot, K, lastO,
                                                                 outpts, Ctot, co);
      co += lastO;
    }
  };

  // ---- SA1: N=2048, no features, npoint=512 ----
  static const float r1[3] = {0.1f, 0.2f, 0.4f};
  static const int   k1[3] = {16, 32, 128};
  static const int   m1[3][3] = {{32,32,64},{64,64,128},{64,96,128}};
  run_sa(xyz_t, 2048, nullptr, 0, 512, r1, k1, m1, /*pbase=*/1, fidx1, nxyz1, l1pts, 320);

  // ---- SA2: N=512, Cf=320, npoint=128 ----
  static const float r2a[3] = {0.2f, 0.4f, 0.8f};
  static const int   k2[3] = {32, 64, 128};
  static const int   m2[3][3] = {{64,64,128},{128,128,256},{128,128,256}};
  run_sa(nxyz1, 512, l1pts, 320, 128, r2a, k2, m2, /*pbase=*/37, fidx2, nxyz2, l2pts, 640);

  // ---- SA3: group-all over 128 points, C=3+640=643 (stride 644) ----
  build_sa3_k<<<cdiv(1024,256),256,0,stream>>>(nxyz2, l2pts, gbuf, 1024);
  static const int m3[3] = {256, 512, 1024};
  float* last = nullptr; int lastO = 0;
  run_layers(gbuf, 1024, 643, m3, /*pbase=*/73, &last, &lastO);
  maxpool_k<<<cdiv(8*1024,256),256,0,stream>>>(last, 8, 128, 1024, xpool, 1024, 0);

  // ---- FC head: d_out = [logits(8*40) | feat(8*512)] ----
  float* logits = (float*)d_out;
  float* feat   = (float*)d_out + 8*40;
  fc_k<<<cdiv(8*512,256),256,0,stream>>>(xpool, P(85), P(86), feat, 8, 1024, 512);
  bn1d_relu_k<<<cdiv(512,256),256,0,stream>>>(feat, h1, 8, 512, P(87), P(88));
  fc_k<<<cdiv(8*256,256),256,0,stream>>>(h1, P(89), P(90), t2, 8, 512, 256);
  bn1d_relu_k<<<cdiv(256,256),256,0,stream>>>(t2, h2, 8, 256, P(91), P(92));
  fc_k<<<cdiv(8*40,256),256,0,stream>>>(h2, P(93), P(94), logits, 8, 256, 40);
}